// JointsOHKMCoorLoss_51221779972396
// MI455X (gfx1250) — compile-verified
//
#include <hip/hip_runtime.h>
#include <hip/hip_bf16.h>

// ---------------------------------------------------------------------------
// JointsOHKMMSELoss for MI455X (gfx1250, wave32).
//
// Phase 1 (memory bound, ~214 MB @ 23.3 TB/s ~= 9.2 us floor):
//   per-row weighted sum-of-squared-diff over D=3072, 16 rows per block,
//   16 threads/row, float4 nontemporal streams, WMMA f32 16x16x4 (B = ones)
//   performs the 256-partial -> 16-row-sum reduction in 4 chained matrix ops.
// Phase 2 (trivial): per-batch top-8-of-17 selection + mean, one block.
// ---------------------------------------------------------------------------

typedef __attribute__((ext_vector_type(4))) float f4;
typedef __attribute__((ext_vector_type(2))) float v2f;
typedef __attribute__((ext_vector_type(8))) float v8f;

#define B_DIM 512
#define K_DIM 17
#define D_DIM 3072
#define ROWS (B_DIM * K_DIM)          // 8704 (b,k) rows
#define ROWS_PER_BLOCK 16
#define NBLK (ROWS / ROWS_PER_BLOCK)  // 544 exact
#define TPR 16                        // threads per row
#define F4_PER_THREAD (D_DIM / (TPR * 4))  // 48 float4 loads per stream
#define TOPK 8

__global__ __launch_bounds__(256) void ohkm_rowloss_kernel(
    const float* __restrict__ outp, const float* __restrict__ targ,
    const float* __restrict__ tw, float* __restrict__ loss) {
  // 16 rows x 16 partials, stride 17 floats to avoid LDS bank conflicts.
  __shared__ float parts[ROWS_PER_BLOCK * 17];
  __shared__ float sums[ROWS_PER_BLOCK];

  const int tid = threadIdx.x;
  const int row = tid >> 4;   // 0..15
  const int j   = tid & 15;   // 0..15 (column group within row)
  const long rowg = (long)blockIdx.x * ROWS_PER_BLOCK + row;

  const f4* __restrict__ po = (const f4*)(outp + rowg * (long)D_DIM) + j;
  const f4* __restrict__ pt = (const f4*)(targ + rowg * (long)D_DIM) + j;

  // Streaming squared-diff accumulation: 48 x (2 x 16B NT loads) per thread.
  // Half-wave (16 lanes) covers 256 contiguous bytes of one row per iter.
  float acc = 0.0f;
#pragma unroll 8
  for (int i = 0; i < F4_PER_THREAD; ++i) {
    f4 o = __builtin_nontemporal_load(po + i * TPR);
    f4 t = __builtin_nontemporal_load(pt + i * TPR);
    f4 d = o - t;
    acc = fmaf(d.x, d.x, acc);
    acc = fmaf(d.y, d.y, acc);
    acc = fmaf(d.z, d.z, acc);
    acc = fmaf(d.w, d.w, acc);
  }
  parts[row * 17 + j] = acc;
  __syncthreads();

  // Wave 0 (EXEC all-ones for its 32 lanes) reduces the 16x16 partial tile:
  // D = A(16x4) x ones(4x16) + C, chained over 4 column chunks.
  // A layout (32-bit, 16x4): lanes 0-15 hold K=0,1 in v0,v1; lanes 16-31 K=2,3.
  if (tid < 32) {
    const int m  = tid & 15;
    const int kh = tid >> 4;  // 0: K=0,1   1: K=2,3
    v8f c = {};
    v2f b;
    b[0] = 1.0f;
    b[1] = 1.0f;
#pragma unroll
    for (int cc = 0; cc < 4; ++cc) {
      v2f a;
      a[0] = parts[m * 17 + cc * 4 + kh * 2 + 0];
      a[1] = parts[m * 17 + cc * 4 + kh * 2 + 1];
      // (neg_a, A, neg_b, B, c_mod, C, reuse_a, reuse_b)
      c = __builtin_amdgcn_wmma_f32_16x16x4_f32(
          false, a, false, b, (short)0, c, false, false);
    }
    // D layout: lanes 0-15 vgpr r -> M=r; lanes 16-31 vgpr r -> M=8+r.
    if (m == 0) {
#pragma unroll
      for (int r = 0; r < 8; ++r) sums[kh * 8 + r] = c[r];
    }
  }
  __syncthreads();

  if (tid < ROWS_PER_BLOCK) {
    const long rg = (long)blockIdx.x * ROWS_PER_BLOCK + tid;
    loss[rg] = sums[tid] * tw[rg];
  }
}

__global__ __launch_bounds__(256) void ohkm_topk_mean_kernel(
    const float* __restrict__ loss, float* __restrict__ out) {
  __shared__ float red[256];
  const int tid = threadIdx.x;

  float local = 0.0f;
#pragma unroll
  for (int rep = 0; rep < 2; ++rep) {
    const int b = tid + rep * 256;  // 2 batches per thread, 512 total
    float v[K_DIM];
#pragma unroll
    for (int k = 0; k < K_DIM; ++k) v[k] = loss[b * K_DIM + k];

    // Branchless selection of top-8: fully unrolled -> v_cndmask chains,
    // keeps v[] in registers (no scratch).
    float s = 0.0f;
#pragma unroll
    for (int it = 0; it < TOPK; ++it) {
      float mx = v[0];
      int mi = 0;
#pragma unroll
      for (int k = 1; k < K_DIM; ++k) {
        bool g = v[k] > mx;
        mx = g ? v[k] : mx;
        mi = g ? k : mi;
      }
      s += mx;
      const float ninf = -__builtin_inff();
#pragma unroll
      for (int k = 0; k < K_DIM; ++k) v[k] = (k == mi) ? ninf : v[k];
    }
    local += s * (1.0f / (float)TOPK);
  }

  red[tid] = local;
  __syncthreads();
#pragma unroll
  for (int s = 128; s > 0; s >>= 1) {
    if (tid < s) red[tid] += red[tid + s];
    __syncthreads();
  }
  if (tid == 0) out[0] = red[0] * (1.0f / (float)B_DIM);
}

extern "C" void kernel_launch(void* const* d_in, const int* in_sizes, int n_in,
                              void* d_out, int out_size, void* d_ws,
                              size_t ws_size, hipStream_t stream) {
  (void)in_sizes; (void)n_in; (void)out_size; (void)ws_size;
  const float* outp = (const float*)d_in[0];   // [512,17,3072] f32
  const float* targ = (const float*)d_in[1];   // [512,17,3072] f32
  const float* tw   = (const float*)d_in[2];   // [512,17,1]    f32
  float* loss = (float*)d_ws;                  // 8704 f32 = 34816 B scratch
  float* out  = (float*)d_out;                 // scalar f32

  ohkm_rowloss_kernel<<<NBLK, 256, 0, stream>>>(outp, targ, tw, loss);
  ohkm_topk_mean_kernel<<<1, 256, 0, stream>>>(loss, out);
}